// RelativePositionalEncoder_64218351010458
// MI455X (gfx1250) — compile-verified
//
#include <hip/hip_runtime.h>

typedef float v2f __attribute__((ext_vector_type(2)));
typedef float v8f __attribute__((ext_vector_type(8)));

#define M_DIM   8
#define D_DIM   2048
#define K_DIM   64          // f*c
#define M_TOT   (M_DIM * D_DIM)   // 16384 rows
#define NT      257         // distinct relative-distance indices
#define NPAD    272         // 17 tiles of 16, >= 257; pe has 4095 rows so reads are in-bounds
#define NTILES  17

// -------- Phase 1: V[row, n] = sum_q A[row, q] * pe[n, q]  (16384 x 272, K=64) --------
// One wave32 per 16x16 output tile; V_WMMA_F32_16X16X4_F32, 16 K-steps of K=4.
__global__ void __launch_bounds__(32)
relpe_gemm_kernel(const float* __restrict__ A, const float* __restrict__ pe,
                  float* __restrict__ V) {
    const int tile = blockIdx.x;
    const int nt   = tile % NTILES;
    const int mt   = tile / NTILES;
    const int row0 = mt * 16;
    const int n0   = nt * 16;

    const int lane = threadIdx.x;      // 0..31
    const int l15  = lane & 15;
    const int grp  = lane >> 4;        // 0 or 1

    // A (16x4 f32) fragment: lane l15 = row M, group selects K-pair {2g, 2g+1}
    const float* arow = A  + (size_t)(row0 + l15) * K_DIM + 2 * grp;
    // B (4x16 f32) fragment: B[k][n] = pe[n][k]; lane l15 = col N, group selects K-pair
    const float* brow = pe + (size_t)(n0  + l15) * K_DIM + 2 * grp;

    v8f c = {};
    #pragma unroll
    for (int ks = 0; ks < 16; ++ks) {
        v2f a = *(const v2f*)(arow + ks * 4);
        v2f b = *(const v2f*)(brow + ks * 4);
        // 8 args: (neg_a, A, neg_b, B, c_mod, C, reuse_a, reuse_b)
        c = __builtin_amdgcn_wmma_f32_16x16x4_f32(false, a, false, b,
                                                  (short)0, c, false, false);
    }

    // C/D 16x16 f32 layout: VGPR r holds row (r + 8*grp), col = l15
    float* vout = V + n0 + l15;
    #pragma unroll
    for (int r = 0; r < 8; ++r) {
        vout[(size_t)(row0 + r + 8 * grp) * NPAD] = c[r];
    }
}

// -------- Phase 2: out[b,i,j] = V[b*D+i, clip(i-j,-128,128)+128] --------
// One block per (b,i) row. Stage the 257 V values in LDS, then coalesced float4 stores.
__global__ void __launch_bounds__(256)
relpe_scatter_kernel(const float* __restrict__ V, float* __restrict__ out) {
    __shared__ float lds[NT];
    const int rowid = blockIdx.x;           // b*D + i
    const int i     = rowid & (D_DIM - 1);
    const int tid   = threadIdx.x;

    for (int t = tid; t < NT; t += 256)
        lds[t] = V[(size_t)rowid * NPAD + t];
    __syncthreads();

    float* orow = out + (size_t)rowid * D_DIM;
    const int j0 = tid * 8;                 // 256 threads * 8 = 2048 cols
    float4 w0, w1;
    #pragma unroll
    for (int u = 0; u < 8; ++u) {
        int t = i - (j0 + u);
        t = (t < -128) ? -128 : ((t > 128) ? 128 : t);
        float v = lds[t + 128];
        if (u < 4) (&w0.x)[u] = v; else (&w1.x)[u - 4] = v;
    }
    *(float4*)(orow + j0)     = w0;
    *(float4*)(orow + j0 + 4) = w1;
}

// -------- Fallback (only if workspace is too small): direct per-element dot --------
__global__ void __launch_bounds__(256)
relpe_direct_kernel(const float* __restrict__ A, const float* __restrict__ pe,
                    float* __restrict__ out) {
    const size_t gid = (size_t)blockIdx.x * 256 + threadIdx.x;   // b*D*D + i*D + j
    const int j   = (int)(gid & (D_DIM - 1));
    const int bi  = (int)(gid >> 11);        // b*D + i
    const int i   = bi & (D_DIM - 1);
    int t = i - j;
    t = (t < -128) ? -128 : ((t > 128) ? 128 : t);
    const float* arow = A  + (size_t)bi * K_DIM;
    const float* prow = pe + (size_t)(t + 128) * K_DIM;
    float acc = 0.f;
    #pragma unroll 8
    for (int q = 0; q < K_DIM; ++q) acc += arow[q] * prow[q];
    out[gid] = acc;
}

extern "C" void kernel_launch(void* const* d_in, const int* in_sizes, int n_in,
                              void* d_out, int out_size, void* d_ws, size_t ws_size,
                              hipStream_t stream) {
    const float* attn_x = (const float*)d_in[0];   // (8, 2048, 8, 8) fp32 == (16384, 64)
    const float* pe     = (const float*)d_in[1];   // (4095, 64) fp32
    float*       out    = (float*)d_out;           // (8, 2048, 1, 2048, 1) fp32

    const size_t need = (size_t)M_TOT * NPAD * sizeof(float);   // ~17.8 MB
    if (ws_size >= need) {
        float* V = (float*)d_ws;
        relpe_gemm_kernel<<<dim3((M_TOT / 16) * NTILES), dim3(32), 0, stream>>>(attn_x, pe, V);
        relpe_scatter_kernel<<<dim3(M_TOT), dim3(256), 0, stream>>>(V, out);
    } else {
        const size_t total = (size_t)M_DIM * D_DIM * D_DIM;
        relpe_direct_kernel<<<dim3((unsigned)(total / 256)), dim3(256), 0, stream>>>(attn_x, pe, out);
    }
}